// SignatureTokenEmbedding_22393959481676
// MI455X (gfx1250) — compile-verified
//
#include <hip/hip_runtime.h>

typedef __attribute__((ext_vector_type(2))) float v2f;
typedef __attribute__((ext_vector_type(8))) float v8f;

// Explicit global (address_space(1)) pointee types so selected pointers
// lower to global_load/global_store instead of flat_* (flat costs both
// LOADcnt+DScnt and contends with the LDS pipe used by ds_bpermute).
typedef const __attribute__((address_space(1))) float gc_float;
typedef const __attribute__((address_space(1))) v2f   gc_v2f;
typedef __attribute__((address_space(1))) float       g_float;
typedef __attribute__((address_space(1))) v2f         g_v2f;

#define NTOK   (16 * 4096)   // 65536 tokens
#define NEMBD  512
#define MAXVD  128
#define NTYPE  6
#define MAX_TILES 4104       // sum_t ceil(cnt_t/16) <= 4096 + 5

// ---------------------------------------------------------------------------
// Pass 1: histogram of token types.  ws[0..5] = counts
// ---------------------------------------------------------------------------
__global__ void count_types_kernel(const int* __restrict__ tt,
                                   int* __restrict__ ws, int n) {
  int i = blockIdx.x * blockDim.x + threadIdx.x;
  if (i < n) {
    int t = tt[i];
    t = t > 5 ? 5 : (t < 0 ? 0 : t);
    atomicAdd(&ws[t], 1);
  }
}

// ---------------------------------------------------------------------------
// Pass 2: exclusive scan of 6 counters.  ws[6..11] = segment bases
// ---------------------------------------------------------------------------
__global__ void scan_types_kernel(int* __restrict__ ws) {
  if (threadIdx.x == 0) {
    int acc = 0;
    for (int t = 0; t < NTYPE; ++t) { ws[6 + t] = acc; acc += ws[t]; }
  }
}

// ---------------------------------------------------------------------------
// Pass 3: group token indices by type.  ws[12..17] = cursors, perm[NTOK]
// ---------------------------------------------------------------------------
__global__ void build_perm_kernel(const int* __restrict__ tt,
                                  int* __restrict__ ws,
                                  int* __restrict__ perm, int n) {
  int i = blockIdx.x * blockDim.x + threadIdx.x;
  if (i < n) {
    int t = tt[i];
    t = t > 5 ? 5 : (t < 0 ? 0 : t);
    int pos = ws[6 + t] + atomicAdd(&ws[12 + t], 1);
    perm[pos] = i;
  }
}

// ---------------------------------------------------------------------------
// Fully-unrolled WMMA K-loop, specialized on the (compile-time) inner dim D.
// All loads unconditional v2f global_load_b64 (8B aligned: D % 16 == 0,
// token rows are 512B aligned).
// ---------------------------------------------------------------------------
template <int D>
__device__ __forceinline__ void wmma_tile_body(
    gc_float* __restrict__ Wt, gc_float* __restrict__ vrow,
    int nb, int nIdx, int khalf,
    v8f& c0, v8f& c1, v8f& c2, v8f& c3)
{
  gc_v2f* __restrict__ av = (gc_v2f*)vrow;
  gc_v2f* __restrict__ wv = (gc_v2f*)(Wt + (size_t)(nb + nIdx) * D);
  constexpr int RS = (16 * D) / 2;   // 16 weight rows, in v2f units
#pragma unroll
  for (int k0 = 0; k0 < D; k0 += 4) {
    const int kh = (k0 >> 1) + khalf;  // this lane's K pair
    v2f a   = av[kh];
    v2f b0v = wv[kh];
    v2f b1v = wv[kh + RS];
    v2f b2v = wv[kh + 2 * RS];
    v2f b3v = wv[kh + 3 * RS];
    c0 = __builtin_amdgcn_wmma_f32_16x16x4_f32(false, a, false, b0v, (short)0, c0, false, false);
    c1 = __builtin_amdgcn_wmma_f32_16x16x4_f32(false, a, false, b1v, (short)0, c1, false, false);
    c2 = __builtin_amdgcn_wmma_f32_16x16x4_f32(false, a, false, b2v, (short)0, c2, false, false);
    c3 = __builtin_amdgcn_wmma_f32_16x16x4_f32(false, a, false, b3v, (short)0, c3, false, false);
  }
}

// ---------------------------------------------------------------------------
// Pass 4: grouped GEMM.  Block = 16-token tile (one type) x 512 columns.
// 8 waves/block, each wave: 16x64 output via V_WMMA_F32_16X16X4_F32.
// ---------------------------------------------------------------------------
__global__ __launch_bounds__(256) void fused_embed_gemm_kernel(
    const int* __restrict__ ws,            // [0..5]=counts, [6..11]=bases
    const int* __restrict__ perm,
    const float* __restrict__ token_value, // [NTOK, 128]
    const float* __restrict__ type_emb,    // [6, 512]
    const float* __restrict__ w0, const float* __restrict__ bias0,
    const float* __restrict__ w1, const float* __restrict__ bias1,
    const float* __restrict__ w2, const float* __restrict__ bias2,
    const float* __restrict__ w3, const float* __restrict__ bias3,
    const float* __restrict__ w4, const float* __restrict__ bias4,
    const float* __restrict__ w5, const float* __restrict__ bias5,
    float* __restrict__ out)               // [NTOK, 512]
{
  gc_float* W[NTYPE]  = {(gc_float*)w0, (gc_float*)w1, (gc_float*)w2,
                         (gc_float*)w3, (gc_float*)w4, (gc_float*)w5};
  gc_float* BS[NTYPE] = {(gc_float*)bias0, (gc_float*)bias1, (gc_float*)bias2,
                         (gc_float*)bias3, (gc_float*)bias4, (gc_float*)bias5};

  // Map flat tile id -> (type segment, row tile).  Uniform per block.
  int tid = blockIdx.x;
  int type = -1, rowTile = 0, rows = 0, segBase = 0, acc = 0;
#pragma unroll
  for (int t = 0; t < NTYPE; ++t) {
    int cnt = ws[t];
    int nt  = (cnt + 15) >> 4;
    if (type < 0 && tid < acc + nt) {
      type = t; rowTile = tid - acc; rows = cnt; segBase = ws[6 + t];
    }
    acc += nt;
  }
  if (type < 0) return;   // whole block exits together

  gc_float* __restrict__ Wt = W[type];
  gc_float* __restrict__ bt = BS[type];
  gc_float* __restrict__ te = (gc_float*)type_emb + (size_t)type * NEMBD;
  g_float*  __restrict__ og = (g_float*)out;

  // ---- rtg (d == 1): rank-1 update on the VALU path (uniform branch) ----
  if (type == 5) {
    const int col = (int)threadIdx.x * 2;     // 256 threads x 2 cols = 512
    const float wv0 = Wt[col], wv1 = Wt[col + 1];
    v2f addv;
    addv.x = te[col]     + bt[col];
    addv.y = te[col + 1] + bt[col + 1];
#pragma unroll
    for (int m = 0; m < 16; ++m) {
      const int rr = rowTile * 16 + m;
      if (rr < rows) {
        const int tok = perm[segBase + rr];
        const float v0 = token_value[(size_t)tok * MAXVD];
        v2f o;
        o.x = fmaf(v0, wv0, addv.x);
        o.y = fmaf(v0, wv1, addv.y);
        *(g_v2f*)(og + (size_t)tok * NEMBD + col) = o;
      }
    }
    return;
  }

  // ---- WMMA path: D in {32,16,128,64,96}, all multiples of 16 ----
  const int lane  = threadIdx.x & 31;
  const int wave  = threadIdx.x >> 5;   // 0..7 -> 64 output cols each
  const int nb    = wave * 64;
  const int nIdx  = lane & 15;
  const int khalf = lane >> 4;          // 0: K={0,1}, 1: K={2,3}

  // A-row gather: clamp instead of guard; garbage rows are never stored,
  // and each output row depends only on its own A row.  rows >= 1 here.
  const int  mrow  = rowTile * 16 + nIdx;
  const int  mrowc = mrow < rows ? mrow : rows - 1;
  const int  tokA  = perm[segBase + mrowc];
  gc_float* __restrict__ vrow = (gc_float*)token_value + (size_t)tokA * MAXVD;

  v8f c0 = {}; v8f c1 = {}; v8f c2 = {}; v8f c3 = {};

  switch (type) {  // block-uniform dispatch to compile-time K loops
    case 0: wmma_tile_body<32> (Wt, vrow, nb, nIdx, khalf, c0, c1, c2, c3); break;
    case 1: wmma_tile_body<16> (Wt, vrow, nb, nIdx, khalf, c0, c1, c2, c3); break;
    case 2: wmma_tile_body<128>(Wt, vrow, nb, nIdx, khalf, c0, c1, c2, c3); break;
    case 3: wmma_tile_body<64> (Wt, vrow, nb, nIdx, khalf, c0, c1, c2, c3); break;
    default: wmma_tile_body<96>(Wt, vrow, nb, nIdx, khalf, c0, c1, c2, c3); break;
  }

  // Epilogue: type embedding + bias, uniform over the tile's rows.
  const int   n0   = nb + nIdx;
  const float add0 = te[n0]      + bt[n0];
  const float add1 = te[n0 + 16] + bt[n0 + 16];
  const float add2 = te[n0 + 32] + bt[n0 + 32];
  const float add3 = te[n0 + 48] + bt[n0 + 48];

  // C/D layout: VGPR r -> M=r (lanes 0-15) / M=r+8 (lanes 16-31).
  // Row tokens come from lane (mbase+r)'s A-phase perm value via shuffle.
  const int mbase = khalf * 8;
#pragma unroll
  for (int r = 0; r < 8; ++r) {
    const int rr  = rowTile * 16 + mbase + r;
    const int tok = __shfl(tokA, mbase + r, 32);   // ds_bpermute, full EXEC
    if (rr < rows) {
      g_float* __restrict__ orow = og + (size_t)tok * NEMBD;
      orow[n0]      = c0[r] + add0;
      orow[n0 + 16] = c1[r] + add1;
      orow[n0 + 32] = c2[r] + add2;
      orow[n0 + 48] = c3[r] + add3;
    }
  }
}

// ---------------------------------------------------------------------------
extern "C" void kernel_launch(void* const* d_in, const int* in_sizes, int n_in,
                              void* d_out, int out_size, void* d_ws, size_t ws_size,
                              hipStream_t stream) {
  const int*   token_type  = (const int*)d_in[0];
  // d_in[1] token_time, d_in[2] token_group: unused by the reference math
  const float* token_value = (const float*)d_in[3];
  const float* type_emb    = (const float*)d_in[4];
  const float* w0 = (const float*)d_in[5],  *b0 = (const float*)d_in[6];
  const float* w1 = (const float*)d_in[7],  *b1 = (const float*)d_in[8];
  const float* w2 = (const float*)d_in[9],  *b2 = (const float*)d_in[10];
  const float* w3 = (const float*)d_in[11], *b3 = (const float*)d_in[12];
  const float* w4 = (const float*)d_in[13], *b4 = (const float*)d_in[14];
  const float* w5 = (const float*)d_in[15], *b5 = (const float*)d_in[16];
  float* out = (float*)d_out;

  int* wsi  = (int*)d_ws;       // [0..5] counts, [6..11] bases, [12..17] cursors
  int* perm = wsi + 32;         // [NTOK]

  hipMemsetAsync(wsi, 0, 32 * sizeof(int), stream);
  count_types_kernel<<<NTOK / 256, 256, 0, stream>>>(token_type, wsi, NTOK);
  scan_types_kernel<<<1, 32, 0, stream>>>(wsi);
  build_perm_kernel<<<NTOK / 256, 256, 0, stream>>>(token_type, wsi, perm, NTOK);
  fused_embed_gemm_kernel<<<MAX_TILES, 256, 0, stream>>>(
      wsi, perm, token_value, type_emb,
      w0, b0, w1, b1, w2, b2, w3, b3, w4, b4, w5, b5, out);
}